// LeafSeparationNetwork_60876866453858
// MI455X (gfx1250) — compile-verified
//
#include <hip/hip_runtime.h>

typedef __attribute__((ext_vector_type(2))) float v2f;
typedef __attribute__((ext_vector_type(8))) float v8f;

#define BATCH 4
#define NPTS  4096
#define FDIM  256
#define KNN   16
#define MIN_LEAF 10
#define SPLIT 4                          // candidate-range split (waves per block)
#define TILES_PER_WAVE (NPTS / 16 / SPLIT)   // 64

// Wave-local LDS ordering: DS ops are in-order within a wave (ISA 7.3), so we
// only need to stop the compiler from reordering and drain DScnt.
#define WAVE_LDS_SYNC() do {                              \
    __builtin_amdgcn_wave_barrier();                      \
    asm volatile("s_wait_dscnt 0x0" ::: "memory");        \
    __builtin_amdgcn_wave_barrier();                      \
} while (0)

// ---------------------------------------------------------------------------
// Kernel 1: per-point squared norms |p|^2 (ranking term; |q|^2 drops out)
// ---------------------------------------------------------------------------
__global__ __launch_bounds__(256)
void sqnorm_kernel(const float* __restrict__ pts, float* __restrict__ sq) {
    int i = blockIdx.x * blockDim.x + threadIdx.x;
    if (i < BATCH * NPTS) {
        float x = pts[i * 3 + 0];
        float y = pts[i * 3 + 1];
        float z = pts[i * 3 + 2];
        sq[i] = x * x + y * y + z * z;
    }
}

// ---------------------------------------------------------------------------
// Kernel 2: leaf counts per batch (gate: count >= MIN_LEAF)
// ---------------------------------------------------------------------------
__global__ __launch_bounds__(256)
void leafcount_kernel(const int* __restrict__ leaf, int* __restrict__ cnt) {
    __shared__ int red[256];
    int b = blockIdx.x;
    int t = threadIdx.x;
    int s = 0;
    for (int i = t; i < NPTS; i += 256) s += (leaf[b * NPTS + i] > 0) ? 1 : 0;
    red[t] = s;
    __syncthreads();
    for (int off = 128; off > 0; off >>= 1) {
        if (t < off) red[t] += red[t + off];
        __syncthreads();
    }
    if (t == 0) cnt[b] = red[0];
}

// ---------------------------------------------------------------------------
// Kernel 3: WMMA-tiled kNN, candidate-split 4 ways for occupancy.
// Block = 128 threads = 4 waves, all owning the same 32 query points.
// Wave w scans candidates [w*1024, (w+1)*1024): per 16-candidate tile,
// two V_WMMA_F32_16X16X4_F32 build the 32x16 dot block; lanes maintain a
// per-query top-16 (wave-private, in LDS). Lanes 0-31 then merge 4x16 -> 16.
// ---------------------------------------------------------------------------
__global__ __launch_bounds__(128)
void knn_kernel(const float* __restrict__ pts, const float* __restrict__ sqg,
                int* __restrict__ nn_idx) {
    __shared__ float dots[SPLIT][32][17];   // stride 17: coprime with 64 banks
    __shared__ float sqt[SPLIT][16];
    __shared__ float topd[SPLIT][KNN][32];  // k-major: lanes on distinct banks
    __shared__ int   topi[SPLIT][KNN][32];

    const int tid  = threadIdx.x;
    const int lane = tid & 31;              // lane within wave (wave32)
    const int wv   = tid >> 5;              // wave id 0..3 = candidate split
    const int blocksPerBatch = NPTS / 32;   // 128
    const int b = blockIdx.x / blocksPerBatch;
    const int qbase = (blockIdx.x % blocksPerBatch) * 32;
    const float* P = pts + (size_t)b * NPTS * 3;

    const int col  = lane & 15;
    const int half = lane >> 4;

    // A tiles: queries as 16x4 (K padded 3->4 with zero).
    // Load all coords, then branchless select (v_cndmask, no exec branching).
    int q0 = qbase + col;
    int q1 = qbase + 16 + col;
    float q0x = P[q0 * 3 + 0], q0y = P[q0 * 3 + 1], q0z = P[q0 * 3 + 2];
    float q1x = P[q1 * 3 + 0], q1y = P[q1 * 3 + 1], q1z = P[q1 * 3 + 2];
    v2f a0 = { half ? q0z : q0x, half ? 0.0f : q0y };
    v2f a1 = { half ? q1z : q1x, half ? 0.0f : q1y };

    // init this wave's top-k state
    for (int k = 0; k < KNN; ++k) { topd[wv][k][lane] = 3.0e38f; topi[wv][k][lane] = 0; }
    float worst = 3.0e38f;
    int   wslot = 0;

    for (int jt = 0; jt < TILES_PER_WAVE; ++jt) {
        const int jbase = (wv * TILES_PER_WAVE + jt) * 16;
        const int c = jbase + col;

        // B tile: candidates as 4x16 (same per-lane split as A), branchless.
        float cx = P[c * 3 + 0], cy = P[c * 3 + 1], cz = P[c * 3 + 2];
        v2f bt = { half ? cz : cx, half ? 0.0f : cy };

        // prefetch next tile's candidate coords (global_prefetch_b8, no counter)
        if (jt + 1 < TILES_PER_WAVE)
            __builtin_prefetch(&P[(c + 16) * 3], 0, 1);

        if (lane < 16) sqt[wv][lane] = sqg[(size_t)b * NPTS + jbase + lane];

        v8f c0 = {0.f, 0.f, 0.f, 0.f, 0.f, 0.f, 0.f, 0.f};
        v8f c1 = {0.f, 0.f, 0.f, 0.f, 0.f, 0.f, 0.f, 0.f};
        c0 = __builtin_amdgcn_wmma_f32_16x16x4_f32(false, a0, false, bt,
                                                   (short)0, c0, false, false);
        c1 = __builtin_amdgcn_wmma_f32_16x16x4_f32(false, a1, false, bt,
                                                   (short)0, c1, false, false);

        // Scatter 16x16 tiles to LDS. C layout: row = v + 8*half, col = lane&15.
        #pragma unroll
        for (int v = 0; v < 8; ++v) {
            int row = v + half * 8;
            dots[wv][row][col]      = c0[v];
            dots[wv][16 + row][col] = c1[v];
        }
        WAVE_LDS_SYNC();                    // wave-private LDS: no block barrier

        // Each lane scans its own query row; rank by |p|^2 - 2*dot.
        for (int j = 0; j < 16; ++j) {
            float score = sqt[wv][j] - 2.0f * dots[wv][lane][j];
            if (score < worst) {
                topd[wv][wslot][lane] = score;
                topi[wv][wslot][lane] = jbase + j;
                float w = topd[wv][0][lane]; int ws = 0;
                #pragma unroll
                for (int k = 1; k < KNN; ++k) {
                    float v2 = topd[wv][k][lane];
                    if (v2 > w) { w = v2; ws = k; }
                }
                worst = w; wslot = ws;
            }
        }
        WAVE_LDS_SYNC();                    // scans done before next scatter
    }

    __syncthreads();                        // all 4 partial lists visible

    // Merge 4 partial top-16 lists -> final top-16, one query per lane (wave 0).
    if (tid < 32) {
        const int l = tid;
        float w = topd[0][0][l]; int ws = 0;
        #pragma unroll
        for (int k = 1; k < KNN; ++k) {
            float v2 = topd[0][k][l];
            if (v2 > w) { w = v2; ws = k; }
        }
        for (int s = 1; s < SPLIT; ++s) {
            for (int k = 0; k < KNN; ++k) {
                float d = topd[s][k][l];
                int   ii = topi[s][k][l];
                if (d < w) {
                    topd[0][ws][l] = d;
                    topi[0][ws][l] = ii;
                    float nw = topd[0][0][l]; int nws = 0;
                    #pragma unroll
                    for (int k2 = 1; k2 < KNN; ++k2) {
                        float v2 = topd[0][k2][l];
                        if (v2 > nw) { nw = v2; nws = k2; }
                    }
                    w = nw; ws = nws;
                }
            }
        }
        int q = qbase + l;
        int* dst = nn_idx + ((size_t)b * NPTS + q) * KNN;
        for (int k = 0; k < KNN; ++k) dst[k] = topi[0][k][l];
    }
}

// ---------------------------------------------------------------------------
// Kernel 4: gather neighbor features, mean, boundary-weighted blend.
// One block (64 threads) per point; float4 lanes -> B128 loads/stores,
// each 1KB feature row = 64 lanes x 16B = two full cachelines per row.
// ---------------------------------------------------------------------------
__global__ __launch_bounds__(64)
void enhance_kernel(const float* __restrict__ feats, const float* __restrict__ bp,
                    const int* __restrict__ leaf, const int* __restrict__ cnt,
                    const int* __restrict__ nn_idx, float* __restrict__ out) {
    __shared__ int idx_s[KNN];
    int pid = blockIdx.x;                   // 0 .. BATCH*NPTS-1
    int b = pid / NPTS;
    int t = threadIdx.x;                    // 0 .. 63 (f4 index)
    if (t < KNN) idx_s[t] = nn_idx[(size_t)pid * KNN + t];
    __syncthreads();

    const float4* fb = (const float4*)(feats + (size_t)b * NPTS * FDIM);
    const int rowq = FDIM / 4;              // 64 float4 per row
    float4 sum = {0.f, 0.f, 0.f, 0.f};
    #pragma unroll
    for (int k = 0; k < KNN; ++k) {
        float4 v = fb[(size_t)idx_s[k] * rowq + t];
        sum.x += v.x; sum.y += v.y; sum.z += v.z; sum.w += v.w;
    }
    const float inv = 1.0f / KNN;
    float4 mean = {sum.x * inv, sum.y * inv, sum.z * inv, sum.w * inv};

    float4 feat = ((const float4*)feats)[(size_t)pid * rowq + t];
    float p = bp[pid];
    bool cond = (leaf[pid] > 0) && (p > 0.5f) && (cnt[b] >= MIN_LEAF);
    float s = 1.0f + p;
    float4 enh = {mean.x + (feat.x - mean.x) * s,
                  mean.y + (feat.y - mean.y) * s,
                  mean.z + (feat.z - mean.z) * s,
                  mean.w + (feat.w - mean.w) * s};
    ((float4*)out)[(size_t)pid * rowq + t] = cond ? enh : feat;
}

// ---------------------------------------------------------------------------
extern "C" void kernel_launch(void* const* d_in, const int* in_sizes, int n_in,
                              void* d_out, int out_size, void* d_ws, size_t ws_size,
                              hipStream_t stream) {
    (void)in_sizes; (void)n_in; (void)out_size; (void)ws_size;

    const float* points = (const float*)d_in[0];   // [B,N,3]
    const float* feats  = (const float*)d_in[1];   // [B,N,F]
    const float* bp     = (const float*)d_in[2];   // [B,N]
    const int*   leaf   = (const int*)d_in[3];     // [B,N]
    float* out = (float*)d_out;                    // [B,N,F]

    // workspace layout: nn_idx | sqnorms | leaf counts  (~1.1 MB)
    char* ws = (char*)d_ws;
    int*   nn_idx = (int*)ws;
    float* sqg    = (float*)(ws + (size_t)BATCH * NPTS * KNN * sizeof(int));
    int*   cnt    = (int*)(ws + (size_t)BATCH * NPTS * KNN * sizeof(int)
                              + (size_t)BATCH * NPTS * sizeof(float));

    sqnorm_kernel   <<<(BATCH * NPTS + 255) / 256, 256, 0, stream>>>(points, sqg);
    leafcount_kernel<<<BATCH, 256, 0, stream>>>(leaf, cnt);
    knn_kernel      <<<BATCH * (NPTS / 32), 128, 0, stream>>>(points, sqg, nn_idx);
    enhance_kernel  <<<BATCH * NPTS, 64, 0, stream>>>(feats, bp, leaf, cnt, nn_idx, out);
}